// ExphormerModel_16853451669980
// MI455X (gfx1250) — compile-verified
//
#include <hip/hip_runtime.h>

typedef __attribute__((ext_vector_type(2))) float v2f;
typedef __attribute__((ext_vector_type(8))) float v8f;

#define NUM_PL 50000
#define NUM_TR 100000
#define NUM_AR 10000
#define HID 64
#define FEAT 128
#define N_NODES (NUM_PL + NUM_TR + NUM_AR)
#define OFF_TR NUM_PL
#define OFF_AR (NUM_PL + NUM_TR)

// ---------------------------------------------------------------- utilities

__global__ void zero_kernel(float* __restrict__ p, int n) {
  int i = blockIdx.x * blockDim.x + threadIdx.x;
  int stride = gridDim.x * blockDim.x;
  for (; i < n; i += stride) p[i] = 0.0f;
}

// x[row] = emb[row] + type_row  (vectorized float4; i indexes row*16 quads)
__global__ void embed_type_kernel(const float* __restrict__ emb,
                                  const float* __restrict__ trow,
                                  float* __restrict__ xout, int rows) {
  int i = blockIdx.x * blockDim.x + threadIdx.x;
  if (i >= rows * 16) return;
  int c4 = (i & 15) * 4;
  float4 e = *(const float4*)(emb + (long long)i * 4);
  float4 t = *(const float4*)(trow + c4);
  float4 o = make_float4(e.x + t.x, e.y + t.y, e.z + t.z, e.w + t.w);
  *(float4*)(xout + (long long)i * 4) = o;
}

// --------------------------------------------------------------- degrees

__global__ void deg_kernel(const int* __restrict__ ia, const int* __restrict__ ib,
                           int offA, int offB, int n, float* __restrict__ deg) {
  int e = blockIdx.x * blockDim.x + threadIdx.x;
  if (e >= n) return;
  // directed edges a->b and b->a each bump the destination's degree
  unsafeAtomicAdd(&deg[ib[e] + offB], 1.0f);
  unsafeAtomicAdd(&deg[ia[e] + offA], 1.0f);
}

__global__ void deg_inv_kernel(float* __restrict__ deg, int n) {
  int i = blockIdx.x * blockDim.x + threadIdx.x;
  if (i < n) deg[i] = 1.0f / fmaxf(deg[i], 1.0f);
}

// --------------------------------------------------------- edge scatter-add
// 16 lanes per edge pair; each lane moves one float4 in both directions.
// Gathers are half-wave coalesced (16 lanes cover one 256B row); scatters are
// global_atomic_add_f32 resolving in L2 (working set < 192MB L2).

__global__ void edge_scatter_kernel(const int* __restrict__ ia, const int* __restrict__ ib,
                                    int offA, int offB, int n,
                                    const float* __restrict__ x, float* __restrict__ agg) {
  long long gid = (long long)blockIdx.x * blockDim.x + threadIdx.x;
  if (gid >= (long long)n * 16) return;
  int e = (int)(gid >> 4);
  int seg = ((int)gid & 15) * 4;
  int a = ia[e] + offA;
  int b = ib[e] + offB;
  float4 va = *(const float4*)(x + (long long)a * HID + seg);
  float4 vb = *(const float4*)(x + (long long)b * HID + seg);
  float* pb = agg + (long long)b * HID + seg;
  float* pa = agg + (long long)a * HID + seg;
  unsafeAtomicAdd(pb + 0, va.x); unsafeAtomicAdd(pb + 1, va.y);
  unsafeAtomicAdd(pb + 2, va.z); unsafeAtomicAdd(pb + 3, va.w);
  unsafeAtomicAdd(pa + 0, vb.x); unsafeAtomicAdd(pa + 1, vb.y);
  unsafeAtomicAdd(pa + 2, vb.z); unsafeAtomicAdd(pa + 3, vb.w);
}

// ----------------------------------------------- track features: WMMA GEMM
// xout[OFF_TR+m] = track_x[m] @ W.T + b + type_emb[1];  M=100000,K=128,N=64
// Per wave: one 16-row tile, 4 column tiles of 16, K unrolled by 4 via
// V_WMMA_F32_16X16X4_F32. Weights staged in LDS (padded stride -> no bank
// conflicts on the strided B reads).

#define TLDSK (FEAT + 2) // 130

__global__ void __launch_bounds__(256)
track_feat_kernel(const float* __restrict__ track_x, const float* __restrict__ W,
                  const float* __restrict__ bvec, const float* __restrict__ type1,
                  float* __restrict__ xout) {
  __shared__ float lw[HID * TLDSK];
  for (int i = threadIdx.x; i < HID * FEAT; i += blockDim.x) {
    int r = i >> 7, c = i & 127;
    lw[r * TLDSK + c] = W[i];
  }
  __syncthreads();

  int wave = threadIdx.x >> 5;          // wave32
  int lane = threadIdx.x & 31;
  int tile = blockIdx.x * 8 + wave;
  if (tile >= NUM_TR / 16) return;      // uniform per wave; EXEC stays all-1s
  int lrow = lane & 15;
  int half = lane >> 4;                 // A/B k-pair selector
  int row  = tile * 16 + lrow;

  v8f acc[4] = {};
  const float* arow = track_x + (long long)row * FEAT;
  for (int k0 = 0; k0 < FEAT; k0 += 4) {
    v2f a = *(const v2f*)(arow + k0 + half * 2);          // A[m][k0..k0+3]
#pragma unroll
    for (int nt = 0; nt < 4; ++nt) {
      v2f bm = *(const v2f*)(&lw[(nt * 16 + lrow) * TLDSK + k0 + half * 2]); // B[k][n]=W[n][k]
      acc[nt] = __builtin_amdgcn_wmma_f32_16x16x4_f32(false, a, false, bm,
                                                      (short)0, acc[nt], false, false);
    }
  }
#pragma unroll
  for (int nt = 0; nt < 4; ++nt) {
    int col = nt * 16 + lrow;
    float bias = bvec[col] + type1[col];
#pragma unroll
    for (int v = 0; v < 8; ++v) {
      int orow = tile * 16 + half * 8 + v;                 // C layout: M = half*8+v
      xout[(long long)(OFF_TR + orow) * HID + col] = acc[nt][v] + bias;
    }
  }
}

// ------------------------------------------- fused SAGE layer update (WMMA)
// x_out = relu( (agg*deg_inv) @ Wl.T + bl + x_in @ Wr.T );  per-row
// independent, so in-place (x_out == x_in) is safe.

#define SLDSK (HID + 2) // 66

__global__ void __launch_bounds__(256)
sage_update_kernel(const float* __restrict__ x_in, const float* __restrict__ agg,
                   const float* __restrict__ deg_inv,
                   const float* __restrict__ Wl, const float* __restrict__ bl,
                   const float* __restrict__ Wr, float* __restrict__ x_out) {
  __shared__ float lwl[HID * SLDSK];
  __shared__ float lwr[HID * SLDSK];
  for (int i = threadIdx.x; i < HID * HID; i += blockDim.x) {
    int r = i >> 6, c = i & 63;
    lwl[r * SLDSK + c] = Wl[i];
    lwr[r * SLDSK + c] = Wr[i];
  }
  __syncthreads();

  int wave = threadIdx.x >> 5;
  int lane = threadIdx.x & 31;
  int tile = blockIdx.x * 8 + wave;     // grid chosen exact: no guard needed
  int lrow = lane & 15;
  int half = lane >> 4;
  int row  = tile * 16 + lrow;
  float ds = deg_inv[row];

  v8f acc[4] = {};
  const float* xr = x_in + (long long)row * HID;
  const float* ar = agg  + (long long)row * HID;
  for (int k0 = 0; k0 < HID; k0 += 4) {
    v2f am = *(const v2f*)(ar + k0 + half * 2);
    am = am * ds;                                          // mean aggregation
    v2f ax = *(const v2f*)(xr + k0 + half * 2);
#pragma unroll
    for (int nt = 0; nt < 4; ++nt) {
      v2f bL = *(const v2f*)(&lwl[(nt * 16 + lrow) * SLDSK + k0 + half * 2]);
      v2f bR = *(const v2f*)(&lwr[(nt * 16 + lrow) * SLDSK + k0 + half * 2]);
      acc[nt] = __builtin_amdgcn_wmma_f32_16x16x4_f32(false, am, false, bL,
                                                      (short)0, acc[nt], false, false);
      acc[nt] = __builtin_amdgcn_wmma_f32_16x16x4_f32(false, ax, false, bR,
                                                      (short)0, acc[nt], false, false);
    }
  }
#pragma unroll
  for (int nt = 0; nt < 4; ++nt) {
    int col = nt * 16 + lrow;
    float bias = bl[col];
#pragma unroll
    for (int v = 0; v < 8; ++v) {
      int orow = tile * 16 + half * 8 + v;
      x_out[(long long)orow * HID + col] = fmaxf(acc[nt][v] + bias, 0.0f);
    }
  }
}

// ------------------------------------------------------------------ driver

extern "C" void kernel_launch(void* const* d_in, const int* in_sizes, int n_in,
                              void* d_out, int out_size, void* d_ws, size_t ws_size,
                              hipStream_t stream) {
  const float* track_x      = (const float*)d_in[0];
  const int*   pl_tr_src    = (const int*)d_in[1];
  const int*   pl_tr_dst    = (const int*)d_in[2];
  const int*   tr_ar_src    = (const int*)d_in[3];
  const int*   tr_ar_dst    = (const int*)d_in[4];
  const float* playlist_emb = (const float*)d_in[5];
  const float* artist_emb   = (const float*)d_in[6];
  const float* track_W      = (const float*)d_in[7];
  const float* track_b      = (const float*)d_in[8];
  const float* type_emb     = (const float*)d_in[9];
  const float* conv_Wl      = (const float*)d_in[10];
  const float* conv_bl      = (const float*)d_in[11];
  const float* conv_Wr      = (const float*)d_in[12];
  const int E1 = in_sizes[1];
  const int E2 = in_sizes[3];

  float* xbuf = (float*)d_ws;                         // N*64
  float* agg  = xbuf + (size_t)N_NODES * HID;         // N*64
  float* deg  = agg  + (size_t)N_NODES * HID;         // N
  float* xout = (float*)d_out;

  // degrees -> 1/max(deg,1)
  zero_kernel<<<(N_NODES + 255) / 256, 256, 0, stream>>>(deg, N_NODES);
  deg_kernel<<<(E1 + 255) / 256, 256, 0, stream>>>(pl_tr_src, pl_tr_dst, 0, OFF_TR, E1, deg);
  deg_kernel<<<(E2 + 255) / 256, 256, 0, stream>>>(tr_ar_src, tr_ar_dst, OFF_TR, OFF_AR, E2, deg);
  deg_inv_kernel<<<(N_NODES + 255) / 256, 256, 0, stream>>>(deg, N_NODES);

  // node feature assembly
  embed_type_kernel<<<(NUM_PL * 16 + 255) / 256, 256, 0, stream>>>(
      playlist_emb, type_emb, xbuf, NUM_PL);
  embed_type_kernel<<<(NUM_AR * 16 + 255) / 256, 256, 0, stream>>>(
      artist_emb, type_emb + 2 * HID, xbuf + (size_t)OFF_AR * HID, NUM_AR);
  track_feat_kernel<<<(NUM_TR / 16 + 7) / 8, 256, 0, stream>>>(
      track_x, track_W, track_b, type_emb + HID, xbuf);

  for (int l = 0; l < 2; ++l) {
    zero_kernel<<<40000, 256, 0, stream>>>(agg, N_NODES * HID);
    edge_scatter_kernel<<<(int)(((long long)E1 * 16 + 255) / 256), 256, 0, stream>>>(
        pl_tr_src, pl_tr_dst, 0, OFF_TR, E1, xbuf, agg);
    edge_scatter_kernel<<<(int)(((long long)E2 * 16 + 255) / 256), 256, 0, stream>>>(
        tr_ar_src, tr_ar_dst, OFF_TR, OFF_AR, E2, xbuf, agg);
    float* dst = (l == 1) ? xout : xbuf;
    sage_update_kernel<<<N_NODES / 16 / 8, 256, 0, stream>>>(
        xbuf, agg, deg,
        conv_Wl + (size_t)l * HID * HID, conv_bl + (size_t)l * HID,
        conv_Wr + (size_t)l * HID * HID, dst);
  }
}